// Metric_Loss_30193620091102
// MI455X (gfx1250) — compile-verified
//
#include <hip/hip_runtime.h>
#include <hip/hip_bf16.h>

typedef __attribute__((ext_vector_type(2))) float v2f;
typedef __attribute__((ext_vector_type(8))) float v8f;

#define DIM 256            // embedding dim (reference fixed)
#define KSTEPS (DIM / 4)   // 64 wmma k-steps of K=4
#define CHUNKS 16          // column chunks (partial rowsums); must be <= 32
#define MARGIN_F 1.0f
#define BSTRIDE 260        // padded LDS row stride (dwords): banks 4*l16+4t+2*half -> conflict-free
#define BBLK 8             // B fragments per software-pipeline block

// ---------------------------------------------------------------------------
// emb[r]      = (r even) ? text[r] : shape[r/2]
// emb[N1 + r] = (r even) ? shape[r/2] : text[r]
__global__ void build_emb_kernel(const float* __restrict__ text,
                                 const float* __restrict__ shape,
                                 float* __restrict__ emb, int N1) {
  int idx = blockIdx.x * blockDim.x + threadIdx.x;       // over N1*DIM
  int r = idx >> 8, k = idx & (DIM - 1);
  float t = text[idx];
  float s = shape[(r >> 1) * DIM + k];
  size_t lo = (size_t)idx, hi = (size_t)N1 * DIM + idx;
  if (r & 1) { emb[lo] = s; emb[hi] = t; }
  else       { emb[lo] = t; emb[hi] = s; }
}

// ---------------------------------------------------------------------------
// sq[r] = sum_k X[r,k]^2 ; one wave per row
__global__ void rowsq_kernel(const float* __restrict__ X, float* __restrict__ sq) {
  int wave = threadIdx.x >> 5, lane = threadIdx.x & 31;
  int row = blockIdx.x * 8 + wave;
  const float4* xr = (const float4*)(X + (size_t)row * DIM);
  float acc = 0.f;
#pragma unroll
  for (int t = lane; t < DIM / 4; t += 32) {
    float4 v = xr[t];
    acc += v.x * v.x + v.y * v.y + v.z * v.z + v.w * v.w;
  }
#pragma unroll
  for (int off = 16; off >= 1; off >>= 1) acc += __shfl_xor(acc, off, 32);
  if (lane == 0) sq[row] = acc;
}

// ---------------------------------------------------------------------------
// Gram + fused epilogue with LDS-shared B tiles.
// Block = 256 thr (8 waves) owns 128 rows; all waves share one 16-col B tile
// staged in LDS per step (8x reuse of every B byte fetched from L2).
// Inner loop is software-pipelined: B fragments are prefetched one 8-step
// block ahead into a double-buffered register array so ds loads stay ~8 deep.
// partial[chunk*N + r] = sum over this chunk's columns of
//   exp(m - sqrt(relu(sq_r + sq_c - 2*G[r,c]))).
// grid = (N/128, CHUNKS), block = 256.
__global__ void __launch_bounds__(256)
gram_rowsum_kernel(const float* __restrict__ X, const float* __restrict__ sq,
                   float* __restrict__ partial, int N) {
  __shared__ float Bt[16 * BSTRIDE];   // 16 cols x 256 K, padded rows (~16.6 KB)
  const int wave = threadIdx.x >> 5;
  const int lane = threadIdx.x & 31;
  const int half = lane >> 4;          // K%4 in {0,1} vs {2,3}; C rows 0-7 vs 8-15
  const int l16  = lane & 15;
  const int rowBase = blockIdx.x * 128 + wave * 16;

  // Preload this wave's A stripe (16 rows x 256 K) into registers.
  // 32-bit A 16x4 layout: v0 = K even, v1 = K odd within this lane-half.
  float2 a[KSTEPS];
  {
    const float* ap = X + (size_t)(rowBase + l16) * DIM + 2 * half;
#pragma unroll
    for (int t = 0; t < KSTEPS; ++t) a[t] = *(const float2*)(ap + 4 * t);
  }
  float sqr[8];
#pragma unroll
  for (int v = 0; v < 8; ++v) sqr[v] = sq[rowBase + v + 8 * half];

  const int tilesPerChunk = (N / 16) / CHUNKS;
  const int tileStart = blockIdx.y * tilesPerChunk;

  float racc[8];
#pragma unroll
  for (int v = 0; v < 8; ++v) racc[v] = 0.f;

  // A 16x256 column tile is contiguous (rows colBase..colBase+15, full K):
  // stage 16KB through registers, double-buffered across iterations.
  float4 stage[4];
  {
    const float4* src = (const float4*)(X + (size_t)tileStart * 16 * DIM);
#pragma unroll
    for (int i = 0; i < 4; ++i) stage[i] = src[threadIdx.x + i * 256];
  }

  for (int t0 = 0; t0 < tilesPerChunk; ++t0) {
    __syncthreads();                       // LDS free (prev compute done)
#pragma unroll
    for (int i = 0; i < 4; ++i) {
      int l = threadIdx.x + i * 256;       // float4 index into 16x256 tile
      *(float4*)&Bt[(l >> 6) * BSTRIDE + (l & 63) * 4] = stage[i];
    }
    if (t0 + 1 < tilesPerChunk) {          // prefetch next tile during compute
      const float4* nsrc =
          (const float4*)(X + (size_t)(tileStart + t0 + 1) * 16 * DIM);
#pragma unroll
      for (int i = 0; i < 4; ++i) stage[i] = nsrc[threadIdx.x + i * 256];
    }
    __syncthreads();                       // LDS ready

    const int colBase = (tileStart + t0) * 16;
    const float* bp = &Bt[l16 * BSTRIDE + 2 * half];

    // Software-pipelined fragment stream: prefetch block blk+1 while the
    // WMMAs of block blk issue; bb[] indices are all unroll-time constants.
    float2 bb[2][BBLK];
#pragma unroll
    for (int i = 0; i < BBLK; ++i) bb[0][i] = *(const float2*)(bp + 4 * i);

    v8f c = {};
#pragma unroll
    for (int blk = 0; blk < KSTEPS / BBLK; ++blk) {
      const int cur = blk & 1, nxt = cur ^ 1;
      if (blk + 1 < KSTEPS / BBLK) {
#pragma unroll
        for (int i = 0; i < BBLK; ++i)
          bb[nxt][i] = *(const float2*)(bp + 4 * ((blk + 1) * BBLK + i));
      }
#pragma unroll
      for (int i = 0; i < BBLK; ++i) {
        const int t = blk * BBLK + i;
        v2f av = {a[t].x, a[t].y};
        v2f bv = {bb[cur][i].x, bb[cur][i].y};
        // D = A(16x4,f32) x B(4x16,f32) + C(16x16,f32)
        c = __builtin_amdgcn_wmma_f32_16x16x4_f32(
            /*neg_a=*/false, av, /*neg_b=*/false, bv,
            /*c_mod=*/(short)0, c, /*reuse_a=*/false, /*reuse_b=*/false);
      }
    }

    const float sqc = sq[colBase + l16];
#pragma unroll
    for (int v = 0; v < 8; ++v) {
      float d2 = sqr[v] + sqc - 2.0f * c[v];
      float dd = d2 > 0.f ? sqrtf(d2) : 0.f;
      racc[v] += __expf(MARGIN_F - dd);
    }
  }

  // Columns live on the 16 lanes of each half; rows are exclusive to this
  // wave, so reduce across the half and store directly.
#pragma unroll
  for (int v = 0; v < 8; ++v) {
    float s = racc[v];
#pragma unroll
    for (int off = 1; off < 16; off <<= 1) s += __shfl_xor(s, off, 32);
    if (l16 == 0)
      partial[(size_t)blockIdx.y * N + rowBase + v + 8 * half] = s;
  }
}

// ---------------------------------------------------------------------------
// Per pair p (i=2p, j=2p+1): neg = rowsum_i + rowsum_j - 2e^m - 2e^(m-Dij);
// contrib = relu(log(neg) + Dij)^2. One wave per pair, 8 pairs per block.
__global__ void pair_kernel(const float* __restrict__ X, const float* __restrict__ sq,
                            const float* __restrict__ partial,
                            float* __restrict__ blockPartial, int N, int P) {
  __shared__ float lds[8];
  const int wave = threadIdx.x >> 5, lane = threadIdx.x & 31;
  const int p = blockIdx.x * 8 + wave;
  float contrib = 0.f;
  if (p < P) {
    const int i = 2 * p, j = i + 1;
    const float4* xi = (const float4*)(X + (size_t)i * DIM);
    const float4* xj = (const float4*)(X + (size_t)j * DIM);
    float dot = 0.f;
#pragma unroll
    for (int t = lane; t < DIM / 4; t += 32) {
      float4 av = xi[t], bv = xj[t];
      dot += av.x * bv.x + av.y * bv.y + av.z * bv.z + av.w * bv.w;
    }
    float ri = (lane < CHUNKS) ? partial[(size_t)lane * N + i] : 0.f;
    float rj = (lane < CHUNKS) ? partial[(size_t)lane * N + j] : 0.f;
#pragma unroll
    for (int off = 16; off >= 1; off >>= 1) {
      dot += __shfl_xor(dot, off, 32);
      ri  += __shfl_xor(ri,  off, 32);
      rj  += __shfl_xor(rj,  off, 32);
    }
    if (lane == 0) {
      float d2  = sq[i] + sq[j] - 2.f * dot;
      float Dij = d2 > 0.f ? sqrtf(d2) : 0.f;
      float neg = ri + rj - 2.f * __expf(MARGIN_F) - 2.f * __expf(MARGIN_F - Dij);
      float J   = __logf(neg) + Dij;
      float r   = fmaxf(J, 0.f);
      contrib = r * r;
    }
  }
  if (lane == 0) lds[wave] = contrib;
  __syncthreads();
  if (threadIdx.x == 0) {
    float s = 0.f;
#pragma unroll
    for (int w = 0; w < 8; ++w) s += lds[w];
    blockPartial[blockIdx.x] = s;
  }
}

// ---------------------------------------------------------------------------
__global__ void final_kernel(const float* __restrict__ bp_tt, int n_tt, float s_tt,
                             const float* __restrict__ bp_st, int n_st, float s_st,
                             float* __restrict__ out) {
  __shared__ float lds[256];
  float acc = 0.f;
  for (int t = threadIdx.x; t < n_tt; t += 256) acc += bp_tt[t] * s_tt;
  for (int t = threadIdx.x; t < n_st; t += 256) acc += bp_st[t] * s_st;
  lds[threadIdx.x] = acc;
  __syncthreads();
  for (int s = 128; s > 0; s >>= 1) {
    if (threadIdx.x < s) lds[threadIdx.x] += lds[threadIdx.x + s];
    __syncthreads();
  }
  if (threadIdx.x == 0) out[0] = lds[0];
}

// ---------------------------------------------------------------------------
extern "C" void kernel_launch(void* const* d_in, const int* in_sizes, int n_in,
                              void* d_out, int out_size, void* d_ws, size_t ws_size,
                              hipStream_t stream) {
  const float* text  = (const float*)d_in[0];   // [N1, 256]
  const float* shape = (const float*)d_in[1];   // [N1/2, 256]
  const int N1 = in_sizes[0] / DIM;             // 4096
  const int N2 = 2 * N1;                        // 8192
  const int P1 = N1 / 2, P2 = N2 / 2;

  float* ws = (float*)d_ws;
  size_t off = 0;
  float* emb    = ws + off; off += (size_t)N2 * DIM;
  float* sq_t   = ws + off; off += N1;
  float* sq_e   = ws + off; off += N2;
  float* part_t = ws + off; off += (size_t)CHUNKS * N1;
  float* part_e = ws + off; off += (size_t)CHUNKS * N2;
  float* bp_tt  = ws + off; off += P1 / 8;
  float* bp_st  = ws + off; off += P2 / 8;

  build_emb_kernel<<<(N1 * DIM) / 256, 256, 0, stream>>>(text, shape, emb, N1);

  rowsq_kernel<<<N1 / 8, 256, 0, stream>>>(text, sq_t);
  rowsq_kernel<<<N2 / 8, 256, 0, stream>>>(emb,  sq_e);

  gram_rowsum_kernel<<<dim3(N1 / 128, CHUNKS), 256, 0, stream>>>(text, sq_t, part_t, N1);
  gram_rowsum_kernel<<<dim3(N2 / 128, CHUNKS), 256, 0, stream>>>(emb,  sq_e, part_e, N2);

  pair_kernel<<<P1 / 8, 256, 0, stream>>>(text, sq_t, part_t, bp_tt, N1, P1);
  pair_kernel<<<P2 / 8, 256, 0, stream>>>(emb,  sq_e, part_e, bp_st, N2, P2);

  final_kernel<<<1, 256, 0, stream>>>(bp_tt, P1 / 8, 1.f / (2.f * P1),
                                      bp_st, P2 / 8, 1.f / (2.f * P2),
                                      (float*)d_out);
}